// MusicalEmbedding_33715493274183
// MI455X (gfx1250) — compile-verified
//
#include <hip/hip_runtime.h>
#include <cstdint>
#include <cstddef>

// MusicalEmbedding: token-emb gather (504) + type-emb (8) concat -> LayerNorm(512) -> *sqrt(512)
// 65536 rows x 512 f32 out = 128MB NT stores; gathers are L2-resident (4MB table).
// Pure HBM-store-bound: ~6us floor at 23.3 TB/s. No matmul -> WMMA deliberately not used.
//
// CDNA5 specifics:
//  * global_load_async_to_lds_b128 + s_wait_asynccnt (ASYNCcnt) stage gamma/beta per block
//  * global_prefetch_b8 for next row's token-table line
//  * wave32 butterfly reduction; s_barrier_signal/wait split barrier
//  * non-temporal b128 stores for the streamed output

typedef float v4f __attribute__((ext_vector_type(4)));

#define D_MODEL 512
#define D_TOK   504
#define ROWS_PER_BLOCK 8
#define BLOCK_THREADS (ROWS_PER_BLOCK * 32)
#define GRID_BLOCKS 2048

__global__ __launch_bounds__(BLOCK_THREADS) void musical_embed_ln_kernel(
    const long long* __restrict__ x,        // [rows] token ids (int64)
    const float* __restrict__ tok_tab,      // [2048, 504]
    const float* __restrict__ typ_tab,      // [5, 8]
    const float* __restrict__ gamma,        // [512]
    const float* __restrict__ beta,         // [512]
    float* __restrict__ out,                // [rows, 512]
    int n_rows)
{
    __shared__ float sgb[2 * D_MODEL];      // [0..511] = gamma, [512..1023] = beta

    // ---- Stage gamma+beta (4KB) into LDS via gfx1250 async loads (ASYNCcnt) ----
    {
        int t = threadIdx.x;                // 0..255, 16B each -> 4KB total
        const float* src = (t < (D_MODEL / 4)) ? (gamma + t * 4)
                                               : (beta + (t - (D_MODEL / 4)) * 4);
        uint64_t gaddr = (uint64_t)(uintptr_t)src;
        uint32_t laddr = (uint32_t)(uintptr_t)(&sgb[t * 4]);
        asm volatile("global_load_async_to_lds_b128 %0, %1, off"
                     :: "v"(laddr), "v"(gaddr) : "memory");
        asm volatile("s_wait_asynccnt 0" ::: "memory");
    }
    __syncthreads();

    const int wave = threadIdx.x >> 5;
    const int lane = threadIdx.x & 31;
    const float SQ = 22.62741699796952f;    // sqrt(512)

    // Hoist this lane's gamma/beta (pre-scaled by sqrt(512)) into registers:
    // row-invariant, so never touch LDS again in the row loop.
    v4f G[4], B[4];
#pragma unroll
    for (int c = 0; c < 4; ++c) {
        int d = c * 128 + lane * 4;
        v4f g = *(const v4f*)&sgb[d];            // ds_load_b128 (once)
        v4f b = *(const v4f*)&sgb[D_MODEL + d];  // ds_load_b128 (once)
#pragma unroll
        for (int k = 0; k < 4; ++k) { G[c][k] = g[k] * SQ; B[c][k] = b[k] * SQ; }
    }

    const int stride = (int)gridDim.x * ROWS_PER_BLOCK;

    for (int row = (int)blockIdx.x * ROWS_PER_BLOCK + wave; row < n_rows; row += stride) {
        // Prefetch next iteration's token-table row (2KB): each lane covers a
        // 64B-spaced slice -> whole row resident by next iteration.
        int nrow = row + stride;
        if (nrow < n_rows) {
            int ntok = __builtin_amdgcn_readfirstlane((int)x[nrow]);
            __builtin_prefetch(tok_tab + (size_t)ntok * D_TOK + lane * 16, 0, 3);
        }

        // Token id is wave-uniform -> scalarize.
        int tok = __builtin_amdgcn_readfirstlane((int)x[row]);
        // Type bucket: [0,128)->0 [128,256)->1 [256,512)->2 [512,1024)->3 [1024,2048)->4
        int ty = (tok < 128) ? 0 : (tok < 256) ? 1 : (tok < 512) ? 2 : (tok < 1024) ? 3 : 4;

        const float* trow = tok_tab + (size_t)tok * D_TOK;   // 2016B stride, 16B aligned
        const float* yrow = typ_tab + ty * 8;                // 32B stride, 16B aligned

        // Lane L, chunk c covers d = c*128 + L*4 .. +3 -> each load/store is a
        // fully-coalesced 512B burst. 504 boundary is chunk-aligned: only
        // (c==3, lane>=30) chunks redirect to the type row.
        v4f v[4];
#pragma unroll
        for (int c = 0; c < 4; ++c) {
            int d = c * 128 + lane * 4;
            const float* p = (d < D_TOK) ? (trow + d) : (yrow + (d - D_TOK));
            v[c] = *(const v4f*)p;
        }

        // Per-lane partials + 5-step wave32 butterfly.
        float s = 0.f, s2 = 0.f;
#pragma unroll
        for (int c = 0; c < 4; ++c)
#pragma unroll
            for (int k = 0; k < 4; ++k) {
                float e = v[c][k];
                s  += e;
                s2 += e * e;
            }
#pragma unroll
        for (int off = 16; off > 0; off >>= 1) {
            s  += __shfl_xor(s,  off, 32);
            s2 += __shfl_xor(s2, off, 32);
        }

        const float inv  = 1.0f / (float)D_MODEL;
        const float mean = s * inv;
        const float var  = s2 * inv - mean * mean;
        const float rstd = __frsqrt_rn(var + 1e-5f);

        float* orow = out + (size_t)row * D_MODEL;
#pragma unroll
        for (int c = 0; c < 4; ++c) {
            int d = c * 128 + lane * 4;
            v4f r;
#pragma unroll
            for (int k = 0; k < 4; ++k)
                r[k] = ((v[c][k] - mean) * rstd) * G[c][k] + B[c][k];  // sub, mul, fma
            __builtin_nontemporal_store(r, (v4f*)(orow + d));          // NT b128 store
        }
    }
}

extern "C" void kernel_launch(void* const* d_in, const int* in_sizes, int n_in,
                              void* d_out, int out_size, void* d_ws, size_t ws_size,
                              hipStream_t stream) {
    const long long* x   = (const long long*)d_in[0];  // int64 tokens
    const float* tok_tab = (const float*)d_in[1];      // [2048,504] f32
    const float* typ_tab = (const float*)d_in[2];      // [5,8] f32
    const float* gamma   = (const float*)d_in[3];      // [512] f32
    const float* beta    = (const float*)d_in[4];      // [512] f32
    float* out           = (float*)d_out;              // [rows,512] f32

    const int n_rows = in_sizes[0];                    // 16*4096 = 65536
    int blocks = (n_rows + ROWS_PER_BLOCK - 1) / ROWS_PER_BLOCK;
    if (blocks > GRID_BLOCKS) blocks = GRID_BLOCKS;

    musical_embed_ln_kernel<<<blocks, BLOCK_THREADS, 0, stream>>>(
        x, tok_tab, typ_tab, gamma, beta, out, n_rows);
}